// WeightedAtomLayer_5420248727865
// MI455X (gfx1250) — compile-verified
//
#include <hip/hip_runtime.h>
#include <hip/hip_bf16.h>

// CDNA5 (gfx1250) wave32 WMMA types
typedef __attribute__((ext_vector_type(2))) float v2f;
typedef __attribute__((ext_vector_type(8))) float v8f;

#define WAVE_SZ 32

// ---------------------------------------------------------------------------
// Kernel 1: build CSR row offsets from the *sorted* segment_ids.
// offsets[n] = first edge e with seg[e] >= n  (offsets has N+1 entries).
// Each n in [0, N] is written by exactly one thread -> deterministic.
// ---------------------------------------------------------------------------
__global__ void build_offsets_kernel(const unsigned long long* __restrict__ seg,
                                     int E, int N, int* __restrict__ offsets) {
    int e = blockIdx.x * blockDim.x + threadIdx.x;
    if (e >= E) return;
    int s     = (int)seg[e];
    int sprev = (e == 0) ? -1 : (int)seg[e - 1];
    for (int n = sprev + 1; n <= s; ++n) offsets[n] = e;   // boundary writes
    if (e == E - 1) {
        for (int n = s + 1; n <= N; ++n) offsets[n] = E;   // tail (incl. offsets[N])
    }
}

// Branch-free tanh: q = e^(-2|x|) in [0,1]; t = sign(x)*(1-q)/(1+q).
// No overflow (exp arg <= 0), saturates to +-1, abs error ~1e-7.
__device__ __forceinline__ float fast_tanhf(float x) {
    float ax = __builtin_fabsf(x);
    float q  = __expf(-2.0f * ax);                 // v_mul + v_exp_f32
    float t  = (1.0f - q) * __builtin_amdgcn_rcpf(1.0f + q);
    return __builtin_copysignf(t, x);
}

// ---------------------------------------------------------------------------
// Kernel 2: one wave per 16-neuron tile. For each K-tile of 4 edges:
//   B[k, d] = weights[e_k, d] * values[gather_idx[e_k], d]      (4x16 f32)
//   A[m, k] = 1.0 if edge e_k belongs to neuron n0+m else 0.0   (16x4 0/1)
//   C += A x B   via v_wmma_f32_16x16x4_f32  (full f32 precision)
// The 0/1 selection matrix performs the sorted segment-sum inside the MACs;
// C ends up as the 16x16 output tile in the native WMMA C layout.
//
// Lane layouts (ISA 7.12.2, wave32):
//   A 16x4 f32 : lane = m (+16 for K-half h=1); VGPR j holds K = 2h + j
//   B 4x16 f32 : lane = d (+16 for K-half h=1); VGPR j holds K = 2h + j
//   C 16x16 f32: VGPR r holds (M = r + 8h, N = d)
//
// All arrays < 4 GB -> uniform SGPR base + 32-bit byte offsets (saddr form),
// running offsets incremented by constants; tail iteration peeled so the hot
// loop has no clamping.
// ---------------------------------------------------------------------------
__global__ void __launch_bounds__(256)
neuron_tile_wmma_kernel(const float* __restrict__ values,
                        const float* __restrict__ weights,
                        const unsigned* __restrict__ idx_lo,  // low dwords of int64 gather_idx
                        const int* __restrict__ offsets,
                        float* __restrict__ out,
                        int numTiles, int N, int E) {
    const int lane = (int)(threadIdx.x & (WAVE_SZ - 1));
    const int wave = (int)(blockIdx.x * (blockDim.x / WAVE_SZ) + (threadIdx.x / WAVE_SZ));
    if (wave >= numTiles) return;                 // uniform per wave: EXEC stays all-1s

    const int n0 = wave * 16;
    const int h  = lane >> 4;                     // K-half: 0 -> K{0,1}, 1 -> K{2,3}
    const int d  = lane & 15;                     // output dim (B/C column) == A row m

    // Per-row edge range [o_lo, o_hi) for the A selection matrix.
    const int rA = (n0 + d     < N) ? (n0 + d)     : N;
    const int rB = (n0 + d + 1 < N) ? (n0 + d + 1) : N;
    const int o_lo = offsets[rA];
    const int o_hi = offsets[rB];
    const unsigned span = (unsigned)(o_hi - o_lo);     // per-lane row length

    const int sBeg = offsets[n0];                                   // uniform
    const int eEnd = offsets[(n0 + 16 < N) ? (n0 + 16) : N];        // uniform

    const char* __restrict__ wB = (const char*)weights;
    const char* __restrict__ iB = (const char*)idx_lo;
    const char* __restrict__ vB = (const char*)values;
    const unsigned dByte = (unsigned)(d * 4);

    // 32-bit running byte offsets (saddr + voffset addressing).
    unsigned wOff = (unsigned)sBeg * 64u + (unsigned)h * 128u + dByte;  // weights[e0+2h][d]
    unsigned iOff = (unsigned)sBeg * 8u  + (unsigned)h * 16u;           // gather_idx[e0+2h]
    int      rRun = sBeg + 2 * h - o_lo;          // (ea - o_lo), running

    v8f acc = {};
    int e0 = sBeg;

    // Hot loop: full K-tiles of 4 edges, no clamping needed (eEnd <= E).
    for (; e0 + 4 <= eEnd; e0 += 4) {
        __builtin_prefetch(wB + (wOff + 2048u), 0, 1);   // global_prefetch_b8, speculative

        const unsigned ia = *(const unsigned*)(iB + iOff);
        const unsigned ib = *(const unsigned*)(iB + (iOff + 8u));
        const float    wa = *(const float*)(wB + wOff);
        const float    wb = *(const float*)(wB + (wOff + 64u));
        const float    ga = *(const float*)(vB + ((ia << 6) + dByte));  // L2-resident gather
        const float    gb = *(const float*)(vB + ((ib << 6) + dByte));

        v2f Bm; Bm.x = wa * ga; Bm.y = wb * gb;
        v2f Am;
        Am.x = ((unsigned)rRun       < span) ? 1.0f : 0.0f;
        Am.y = ((unsigned)(rRun + 1) < span) ? 1.0f : 0.0f;

        acc = __builtin_amdgcn_wmma_f32_16x16x4_f32(
            false, Am, false, Bm, (short)0, acc, false, false);

        wOff += 256u; iOff += 32u; rRun += 4;
    }

    // Peeled tail (1-3 edges), wave-uniform condition: EXEC stays all-1s.
    if (e0 < eEnd) {
        const int ea  = e0 + 2 * h;
        const int eb  = ea + 1;
        const int eaL = (ea < E - 1) ? ea : (E - 1);   // clamp loads; A zeroes tails
        const int ebL = (eb < E - 1) ? eb : (E - 1);

        const unsigned ia = *(const unsigned*)(iB + (unsigned)eaL * 8u);
        const unsigned ib = *(const unsigned*)(iB + (unsigned)ebL * 8u);
        const float    wa = *(const float*)(wB + ((unsigned)eaL * 64u + dByte));
        const float    wb = *(const float*)(wB + ((unsigned)ebL * 64u + dByte));
        const float    ga = *(const float*)(vB + ((ia << 6) + dByte));
        const float    gb = *(const float*)(vB + ((ib << 6) + dByte));

        v2f Bm; Bm.x = wa * ga; Bm.y = wb * gb;
        v2f Am;
        Am.x = ((unsigned)rRun       < span && ea < eEnd) ? 1.0f : 0.0f;
        Am.y = ((unsigned)(rRun + 1) < span && eb < eEnd) ? 1.0f : 0.0f;

        acc = __builtin_amdgcn_wmma_f32_16x16x4_f32(
            false, Am, false, Bm, (short)0, acc, false, false);
    }

    // Store the 16x16 tile: VGPR r -> output row n0 + r + 8h, column d.
    char* __restrict__ oB = (char*)out;
    const int rowBase = n0 + 8 * h;
#pragma unroll
    for (int r = 0; r < 8; ++r) {
        const int row = rowBase + r;
        if (row < N) {
            *(float*)(oB + ((unsigned)row * 64u + dByte)) = fast_tanhf(acc[r]);
        }
    }
}

// ---------------------------------------------------------------------------
// Harness entry
// ---------------------------------------------------------------------------
extern "C" void kernel_launch(void* const* d_in, const int* in_sizes, int n_in,
                              void* d_out, int out_size, void* d_ws, size_t ws_size,
                              hipStream_t stream) {
    const float*              values  = (const float*)d_in[0];               // [S,16] f32
    const float*              weights = (const float*)d_in[1];               // [E,16] f32
    const unsigned*           idx_lo  = (const unsigned*)d_in[2];            // int64 -> low dwords
    const unsigned long long* seg     = (const unsigned long long*)d_in[3];  // int64, sorted

    const int E = in_sizes[2];        // number of edges
    const int N = out_size / 16;      // number of neurons (output rows)

    int* offsets = (int*)d_ws;        // (N+1) ints of scratch (~2 MB)

    // Phase 1: CSR offsets from sorted segment ids.
    {
        int threads = 256;
        int blocks  = (E + threads - 1) / threads;
        build_offsets_kernel<<<blocks, threads, 0, stream>>>(seg, E, N, offsets);
    }

    // Phase 2: WMMA selection-matrix segmented reduce, one wave per 16 neurons.
    {
        const int numTiles      = (N + 15) / 16;
        const int threads       = 256;                    // 8 waves per block
        const int wavesPerBlock = threads / WAVE_SZ;
        const int blocks        = (numTiles + wavesPerBlock - 1) / wavesPerBlock;
        neuron_tile_wmma_kernel<<<blocks, threads, 0, stream>>>(
            values, weights, idx_lo, offsets, (float*)d_out, numTiles, N, E);
    }
}